// SymmetricNetwork_14379550507104
// MI455X (gfx1250) — compile-verified
//
#include <hip/hip_runtime.h>

typedef __attribute__((ext_vector_type(2))) float v2f;
typedef __attribute__((ext_vector_type(8))) float v8f;

#define AGENTS 50
#define NEI 7
#define GRID 8192
#define HID 64
#define NOUT 2
#define XSTRIDE (32 + 2 * GRID)           // 16416 floats per agent row
#define NSPLIT 16
#define ROWS_PER_BLOCK (GRID / NSPLIT)    // 512
#define WAVES 8
#define TILES_PER_WAVE (ROWS_PER_BLOCK / (WAVES * 16))  // 4

// ---------------------------------------------------------------------------
// Kernel 1: per-agent nonzero-segment counts (gcnt over 8192 grid segments,
// ncnt over 7 neighbor segments).
// ---------------------------------------------------------------------------
__global__ __launch_bounds__(256) void count_kernel(const float* __restrict__ X,
                                                    int* __restrict__ gcnt,
                                                    int* __restrict__ ncnt) {
  const int a = blockIdx.x;
  const int tid = threadIdx.x;
  const float* xr = X + (size_t)a * XSTRIDE;
  const float* xg = xr + 32;

  int cnt = 0;
  for (int s = tid; s < GRID; s += 256) {
    float x0 = xg[2 * s + 0];
    float x1 = xg[2 * s + 1];
    cnt += (x0 != 0.0f || x1 != 0.0f) ? 1 : 0;
  }
  __shared__ int red[256];
  red[tid] = cnt;
  __syncthreads();
  for (int o = 128; o > 0; o >>= 1) {
    if (tid < o) red[tid] += red[tid + o];
    __syncthreads();
  }
  if (tid == 0) gcnt[a] = red[0];
  __syncthreads();

  int nc = 0;
  if (tid < NEI) {
    const float* seg = xr + 4 * tid;
    nc = (seg[0] != 0.0f || seg[1] != 0.0f || seg[2] != 0.0f || seg[3] != 0.0f) ? 1 : 0;
  }
  red[tid] = nc;
  __syncthreads();
  if (tid == 0) {
    int s = 0;
    for (int i = 0; i < NEI; ++i) s += red[i];
    ncnt[a] = s;
  }
}

// ---------------------------------------------------------------------------
// Kernel 2: fused grid branch.
//   h = relu(Xg @ W13 + b13)      (2 -> 64, VALU, built directly as A-frags)
//   g = relu(h  @ W23 + b23)      (64 -> 64, v_wmma_f32_16x16x4_f32)
//   partial[a, split, :] = sum over masked rows of g
// Block = 256 threads (8 waves), grid = (50 agents, 16 splits).
// ---------------------------------------------------------------------------
__global__ __launch_bounds__(256) void grid_branch_kernel(
    const float* __restrict__ X, const float* __restrict__ W13,
    const float* __restrict__ b13, const float* __restrict__ W23,
    const float* __restrict__ b23, const int* __restrict__ gcnt,
    float* __restrict__ gpart) {
  __shared__ float4 wp[HID];         // {W13[0][j], W13[1][j], b13[j], 0}
  __shared__ float w23s[HID * HID];  // 16 KB, row-major [k][j]
  __shared__ float b23s[HID];
  __shared__ float wsacc[WAVES][HID];

  const int a = blockIdx.x;
  const int split = blockIdx.y;
  const int tid = threadIdx.x;

  for (int i = tid; i < HID; i += 256)
    wp[i] = make_float4(W13[i], W13[HID + i], b13[i], 0.0f);
  for (int i = tid; i < HID * HID; i += 256) w23s[i] = W23[i];
  for (int i = tid; i < HID; i += 256) b23s[i] = b23[i];
  __syncthreads();

  const int lane = tid & 31;
  const int wave = tid >> 5;
  const int c = lane >> 4;   // K-half / M-half selector
  const int l15 = lane & 15;
  const int cnt = gcnt[a];
  const float* xg = X + (size_t)a * XSTRIDE + 32;

  float psum[4] = {0.0f, 0.0f, 0.0f, 0.0f};  // per-lane column partials

  for (int t = 0; t < TILES_PER_WAVE; ++t) {
    const int tileBase = split * ROWS_PER_BLOCK + (wave * TILES_PER_WAVE + t) * 16;
    const int row = tileBase + l15;
    const float x0 = xg[2 * row + 0];
    const float x1 = xg[2 * row + 1];

    v8f acc[4] = {};  // four 16x16 N-tiles covering 64 output columns

#pragma unroll
    for (int k = 0; k < 16; ++k) {
      // A-fragment: h[r, 4k + 2c] and h[r, 4k + 2c + 1], computed on the fly.
      const int j0 = 4 * k + 2 * c;
      const float4 w0 = wp[j0];
      const float4 w1 = wp[j0 + 1];
      v2f afrag;
      afrag.x = fmaxf(fmaf(x1, w0.y, fmaf(x0, w0.x, w0.z)), 0.0f);
      afrag.y = fmaxf(fmaf(x1, w1.y, fmaf(x0, w1.x, w1.z)), 0.0f);

#pragma unroll
      for (int n = 0; n < 4; ++n) {
        // B-fragment: W23[4k+2c .. 4k+2c+1, 16n + l15]
        v2f bfrag;
        bfrag.x = w23s[j0 * HID + 16 * n + l15];
        bfrag.y = w23s[(j0 + 1) * HID + 16 * n + l15];
        acc[n] = __builtin_amdgcn_wmma_f32_16x16x4_f32(
            false, afrag, false, bfrag, (short)0, acc[n], false, false);
      }
    }

    // Epilogue: +b23, relu, mask (s < gcnt), reduce over the 8 rows this lane
    // holds. D layout: lane, VGPR v -> M = v + 8*c, N = l15 (per 16-wide tile).
#pragma unroll
    for (int n = 0; n < 4; ++n) {
      const float bb = b23s[16 * n + l15];
#pragma unroll
      for (int v = 0; v < 8; ++v) {
        const int m = v + 8 * c;
        const float mask = (tileBase + m) < cnt ? 1.0f : 0.0f;
        psum[n] += fmaxf(acc[n][v] + bb, 0.0f) * mask;
      }
    }
  }

  // Lanes l and l+16 hold the same output column: pair-reduce, then per-wave
  // LDS slots, then fixed-order block reduction (deterministic, no atomics).
#pragma unroll
  for (int n = 0; n < 4; ++n) {
    const float v = psum[n] + __shfl_xor(psum[n], 16, 32);
    if (c == 0) wsacc[wave][16 * n + l15] = v;
  }
  __syncthreads();
  if (tid < HID) {
    float s = 0.0f;
    for (int w = 0; w < WAVES; ++w) s += wsacc[w][tid];
    gpart[((size_t)a * NSPLIT + split) * HID + tid] = s;
  }
}

// ---------------------------------------------------------------------------
// Kernel 3: reduce the NSPLIT partials per agent (fixed order).
// ---------------------------------------------------------------------------
__global__ __launch_bounds__(64) void grid_reduce_kernel(const float* __restrict__ gpart,
                                                         float* __restrict__ sgrid) {
  const int a = blockIdx.x;
  const int j = threadIdx.x;
  float s = 0.0f;
  for (int p = 0; p < NSPLIT; ++p) s += gpart[((size_t)a * NSPLIT + p) * HID + j];
  sgrid[a * HID + j] = s;
}

// ---------------------------------------------------------------------------
// Kernel 4: neighbor branch, self branch, head. One block (64 threads) per
// agent; all intermediates in LDS. Tiny (<0.1% of total FLOPs).
// ---------------------------------------------------------------------------
__global__ __launch_bounds__(64) void head_kernel(
    const float* __restrict__ X, const float* __restrict__ W11,
    const float* __restrict__ b11, const float* __restrict__ W21,
    const float* __restrict__ b21, const float* __restrict__ W12,
    const float* __restrict__ b12, const float* __restrict__ W22,
    const float* __restrict__ b22, const float* __restrict__ W3,
    const float* __restrict__ b3, const float* __restrict__ W4,
    const float* __restrict__ b4, const float* __restrict__ W5,
    const float* __restrict__ b5, const int* __restrict__ ncnt,
    const float* __restrict__ sgrid, float* __restrict__ out) {
  __shared__ float hseg[NEI * HID];
  __shared__ float cat[3 * HID];
  __shared__ float tmp[HID];
  __shared__ float h4s[HID];

  const int a = blockIdx.x;
  const int j = threadIdx.x;
  const float* xr = X + (size_t)a * XSTRIDE;

  // neighbor layer 1: [7,4] @ [4,64]
  for (int s = 0; s < NEI; ++s) {
    float v = b11[j];
    for (int f = 0; f < 4; ++f) v = fmaf(xr[4 * s + f], W11[f * HID + j], v);
    hseg[s * HID + j] = fmaxf(v, 0.0f);
  }
  // self layer 1
  {
    float v = b12[j];
    for (int f = 0; f < 4; ++f) v = fmaf(xr[28 + f], W12[f * HID + j], v);
    tmp[j] = fmaxf(v, 0.0f);
  }
  __syncthreads();

  // neighbor layer 2 + masked sum
  {
    const int nc = ncnt[a];
    float sumn = 0.0f;
    for (int s = 0; s < NEI; ++s) {
      float v = b21[j];
      for (int k = 0; k < HID; ++k) v = fmaf(hseg[s * HID + k], W21[k * HID + j], v);
      sumn += fmaxf(v, 0.0f) * (s < nc ? 1.0f : 0.0f);
    }
    cat[j] = sumn;
  }
  // self layer 2
  {
    float v = b22[j];
    for (int k = 0; k < HID; ++k) v = fmaf(tmp[k], W22[k * HID + j], v);
    cat[HID + j] = fmaxf(v, 0.0f);
  }
  cat[2 * HID + j] = sgrid[a * HID + j];
  __syncthreads();

  // head: 192 -> 64
  float v3 = b3[j];
  for (int k = 0; k < 3 * HID; ++k) v3 = fmaf(cat[k], W3[k * HID + j], v3);
  __syncthreads();  // tmp reads (above) done before overwrite
  tmp[j] = fmaxf(v3, 0.0f);
  __syncthreads();

  // 64 -> 64
  float v4 = b4[j];
  for (int k = 0; k < HID; ++k) v4 = fmaf(tmp[k], W4[k * HID + j], v4);
  h4s[j] = fmaxf(v4, 0.0f);
  __syncthreads();

  // 64 -> 2
  if (j < NOUT) {
    float o = b5[j];
    for (int k = 0; k < HID; ++k) o = fmaf(h4s[k], W5[k * NOUT + j], o);
    out[a * NOUT + j] = o;
  }
}

// ---------------------------------------------------------------------------
extern "C" void kernel_launch(void* const* d_in, const int* in_sizes, int n_in,
                              void* d_out, int out_size, void* d_ws, size_t ws_size,
                              hipStream_t stream) {
  (void)in_sizes; (void)n_in; (void)out_size; (void)ws_size;
  const float* X   = (const float*)d_in[0];
  const float* W11 = (const float*)d_in[1];
  const float* b11 = (const float*)d_in[2];
  const float* W21 = (const float*)d_in[3];
  const float* b21 = (const float*)d_in[4];
  const float* W12 = (const float*)d_in[5];
  const float* b12 = (const float*)d_in[6];
  const float* W22 = (const float*)d_in[7];
  const float* b22 = (const float*)d_in[8];
  const float* W13 = (const float*)d_in[9];
  const float* b13 = (const float*)d_in[10];
  const float* W23 = (const float*)d_in[11];
  const float* b23 = (const float*)d_in[12];
  const float* W3  = (const float*)d_in[13];
  const float* b3  = (const float*)d_in[14];
  const float* W4  = (const float*)d_in[15];
  const float* b4  = (const float*)d_in[16];
  const float* W5  = (const float*)d_in[17];
  const float* b5  = (const float*)d_in[18];
  float* out = (float*)d_out;

  // Workspace layout
  int* gcnt = (int*)d_ws;                 // 50 (padded 64)
  int* ncnt = gcnt + 64;                  // 50 (padded 64)
  float* gpart = (float*)(ncnt + 64);     // 50 * 16 * 64
  float* sgrid = gpart + AGENTS * NSPLIT * HID;  // 50 * 64

  count_kernel<<<AGENTS, 256, 0, stream>>>(X, gcnt, ncnt);
  grid_branch_kernel<<<dim3(AGENTS, NSPLIT), 256, 0, stream>>>(X, W13, b13, W23,
                                                               b23, gcnt, gpart);
  grid_reduce_kernel<<<AGENTS, HID, 0, stream>>>(gpart, sgrid);
  head_kernel<<<AGENTS, HID, 0, stream>>>(X, W11, b11, W21, b21, W12, b12, W22,
                                          b22, W3, b3, W4, b4, W5, b5, ncnt,
                                          sgrid, out);
}